// ParallelSebaRMSNorm_36318243455324
// MI455X (gfx1250) — compile-verified
//
#include <hip/hip_runtime.h>
#include <stdint.h>
#include <stddef.h>

// ---------------------------------------------------------------------------
// ParallelSebaRMSNorm for MI455X (gfx1250).
//
// Roofline: 4*8192 rows * (4KB in + 4KB out) = 256 MB HBM traffic, ~0.3 GFLOP.
// => pure bandwidth problem (~11 us at 23.3 TB/s). No matrix math: the CDNA5
// feature that pays is the async global->LDS copy engine (ASYNCcnt), used
// here to double-buffer full rows per wave. Both 128 MB streams are marked
// non-temporal (th:TH_LOAD_NT on the async loads, TH_STORE_NT via
// __builtin_nontemporal_store) so single-use data sits in L2's evict-first
// class instead of thrashing the 192 MB L2.
// ---------------------------------------------------------------------------

typedef float v4f __attribute__((ext_vector_type(4)));

#define HIDDEN   1024
#define EPSF     1e-6f
#define DIMROOT  32.0f          // sqrt(1024)
#define WPB      4              // waves per block (wave32)
#define TPB      (WPB * 32)
#define ROW_B    (HIDDEN * 4)   // 4096 bytes per row

// --- kernel 1: mean over layers of weights -> mean_w[HIDDEN] ---------------
__global__ __launch_bounds__(256) void meanw_kernel(const float* __restrict__ w,
                                                    float* __restrict__ mw,
                                                    int layers) {
  const int c = threadIdx.x;                 // 256 threads * 4 cols = 1024
  v4f acc = {0.f, 0.f, 0.f, 0.f};
  for (int l = 0; l < layers; ++l)
    acc += *(const v4f*)(w + (size_t)l * HIDDEN + c * 4);
  acc *= (1.0f / (float)layers);
  *(v4f*)(mw + c * 4) = acc;
}

// --- async copy of one 4KB row into LDS (8 x b128 per lane-group) ----------
// INST_OFFSET is added to BOTH the LDS and the global address (ISA 15.18.3 /
// CDNA5 ISA pseudocode), so one address pair serves all 8 issues.
// th:TH_LOAD_NT: the input stream is single-use -> evict-first in L2.
__device__ __forceinline__ void issue_row_async(uint64_t gaddr, uint32_t lds_byte) {
  asm volatile(
      "global_load_async_to_lds_b128 %0, %1, off th:TH_LOAD_NT\n\t"
      "global_load_async_to_lds_b128 %0, %1, off offset:512 th:TH_LOAD_NT\n\t"
      "global_load_async_to_lds_b128 %0, %1, off offset:1024 th:TH_LOAD_NT\n\t"
      "global_load_async_to_lds_b128 %0, %1, off offset:1536 th:TH_LOAD_NT\n\t"
      "global_load_async_to_lds_b128 %0, %1, off offset:2048 th:TH_LOAD_NT\n\t"
      "global_load_async_to_lds_b128 %0, %1, off offset:2560 th:TH_LOAD_NT\n\t"
      "global_load_async_to_lds_b128 %0, %1, off offset:3072 th:TH_LOAD_NT\n\t"
      "global_load_async_to_lds_b128 %0, %1, off offset:3584 th:TH_LOAD_NT"
      :: "v"(lds_byte), "v"(gaddr) : "memory");
}

__device__ __forceinline__ void wait_async_le8() {
  asm volatile("s_wait_asynccnt 8" ::: "memory");   // prev 8 issues complete (in-order)
}
__device__ __forceinline__ void wait_async_0() {
  asm volatile("s_wait_asynccnt 0" ::: "memory");
}

// --- kernel 2: one wave per row, double-buffered async pipeline ------------
__global__ __launch_bounds__(TPB) void seba_norm_kernel(const float* __restrict__ x,
                                                        const float* __restrict__ mw,
                                                        float* __restrict__ out,
                                                        int nrows) {
  __shared__ float smem[WPB * 2 * HIDDEN];          // 32 KB: 2 x 4KB per wave
  const int lane  = threadIdx.x & 31;
  const int wid   = threadIdx.x >> 5;
  const int gwave = blockIdx.x * WPB + wid;
  const int wstride = gridDim.x * WPB;

  float* bufp0 = smem + wid * 2 * HIDDEN;
  float* bufp1 = bufp0 + HIDDEN;
  // Generic shared-pointer low 32 bits == LDS byte offset (flat aperture map).
  const uint32_t lb0 = (uint32_t)(uintptr_t)bufp0 + (uint32_t)lane * 16u;
  const uint32_t lb1 = lb0 + (uint32_t)ROW_B;

  // mean_w into registers once (4KB, L2/WGP$-resident after first touch).
  v4f w[8];
#pragma unroll
  for (int j = 0; j < 8; ++j)
    w[j] = *(const v4f*)(mw + (j * 32 + lane) * 4);

  int r = gwave;
  if (r < nrows)
    issue_row_async((uint64_t)(uintptr_t)(x + (size_t)r * HIDDEN) + (uint64_t)(lane * 16),
                    lb0);

  int parity = 0;
  while (r < nrows) {
    const int rn = r + wstride;
    const bool has_next = rn < nrows;
    if (has_next)
      issue_row_async((uint64_t)(uintptr_t)(x + (size_t)rn * HIDDEN) + (uint64_t)(lane * 16),
                      parity ? lb0 : lb1);
    if (has_next) wait_async_le8(); else wait_async_0();

    const float* buf = parity ? bufp1 : bufp0;

    // Row from LDS into registers; lane-local abs-max and sum-of-squares.
    v4f v[8];
    float m = 0.f, ss = 0.f;
#pragma unroll
    for (int j = 0; j < 8; ++j) {
      v[j] = *(const v4f*)(buf + (j * 32 + lane) * 4);
      m = fmaxf(m, fabsf(v[j].x)); ss = fmaf(v[j].x, v[j].x, ss);
      m = fmaxf(m, fabsf(v[j].y)); ss = fmaf(v[j].y, v[j].y, ss);
      m = fmaxf(m, fabsf(v[j].z)); ss = fmaf(v[j].z, v[j].z, ss);
      m = fmaxf(m, fabsf(v[j].w)); ss = fmaf(v[j].w, v[j].w, ss);
    }

    // wave32 reduction (5 shuffle-xor steps).
#pragma unroll
    for (int o = 16; o >= 1; o >>= 1) {
      m  = fmaxf(m, __shfl_xor(m, o, 32));
      ss +=          __shfl_xor(ss, o, 32);
    }

    // scale = DIMROOT * (1/max) * rsqrt(sumsq/max^2 + eps)  (== reference math)
    const float maxv = fmaxf(m, EPSF);
    const float inv  = 1.0f / maxv;
    const float sq   = ss * inv * inv;
    const float k    = DIMROOT * inv * __builtin_amdgcn_rsqf(sq + EPSF);

    float* orow = out + (size_t)r * HIDDEN;
#pragma unroll
    for (int j = 0; j < 8; ++j) {
      v4f o4 = v[j] * k * w[j];
      __builtin_nontemporal_store(o4, (v4f*)(orow + (j * 32 + lane) * 4));
    }

    r = rn;
    parity ^= 1;
  }
}

// ---------------------------------------------------------------------------
extern "C" void kernel_launch(void* const* d_in, const int* in_sizes, int n_in,
                              void* d_out, int out_size, void* d_ws, size_t ws_size,
                              hipStream_t stream) {
  const float* x   = (const float*)d_in[0];   // [4, 8192, 1024] f32
  const float* wts = (const float*)d_in[1];   // [20, 1024] f32
  float* mw  = (float*)d_ws;                  // [1024] scratch
  float* out = (float*)d_out;

  const int nrows  = in_sizes[0] / HIDDEN;    // 32768
  const int layers = in_sizes[1] / HIDDEN;    // 20

  meanw_kernel<<<1, 256, 0, stream>>>(wts, mw, layers);

  // 4096 blocks * 4 waves = 16384 waves -> 2 rows/wave (depth-2 pipeline).
  const int blocks = 4096;
  seba_norm_kernel<<<blocks, TPB, 0, stream>>>(x, mw, out, nrows);
}